// GNNtest_82480551952450
// MI455X (gfx1250) — compile-verified
//
#include <hip/hip_runtime.h>
#include <math.h>

#define NT 32768
#define N0 3190
#define E0 51040

typedef __attribute__((ext_vector_type(2))) float v2f;
typedef __attribute__((ext_vector_type(8))) float v8f;

// ---------------- Encoder: x[60,NT,5] -> h[NT,10] ----------------
__global__ __launch_bounds__(256) void k_encoder(
    const float* __restrict__ x, const float* __restrict__ emb_w,
    const float* __restrict__ fc_w, const float* __restrict__ fc_b,
    const float* __restrict__ fc2_w, const float* __restrict__ fc2_b,
    float* __restrict__ hout)
{
    int t = blockIdx.x * 256 + threadIdx.x;
    if (t >= NT) return;

    float ew[25];
#pragma unroll
    for (int i = 0; i < 25; ++i) ew[i] = emb_w[i];

    float acc[5][20];
#pragma unroll
    for (int e = 0; e < 5; ++e)
#pragma unroll
        for (int k = 0; k < 20; ++k) acc[e][k] = 0.0f;

    for (int f = 0; f < 60; ++f) {
        float xc[5];
#pragma unroll
        for (int c = 0; c < 5; ++c) xc[c] = x[(f * NT + t) * 5 + c];
        float ge[5];
#pragma unroll
        for (int e = 0; e < 5; ++e) {
            float s = 0.0f;
#pragma unroll
            for (int c = 0; c < 5; ++c) s += xc[c] * ew[c * 5 + e];
            ge[e] = fmaxf(s, 0.0f);
        }
#pragma unroll
        for (int k = 0; k < 20; ++k) {
            float w = fc_w[k * 60 + f];
#pragma unroll
            for (int e = 0; e < 5; ++e) acc[e][k] += ge[e] * w;
        }
    }
#pragma unroll
    for (int k = 0; k < 20; ++k) {
        float b = fc_b[k];
#pragma unroll
        for (int e = 0; e < 5; ++e) acc[e][k] = fmaxf(acc[e][k] + b, 0.0f);
    }
    float hd[10];
#pragma unroll
    for (int d = 0; d < 10; ++d) hd[d] = fc2_b[d];
#pragma unroll
    for (int k = 0; k < 20; ++k)
#pragma unroll
        for (int e = 0; e < 5; ++e) {
            float v = acc[e][k];
#pragma unroll
            for (int d = 0; d < 10; ++d) hd[d] += fc2_w[d * 100 + k * 5 + e] * v;
        }
#pragma unroll
    for (int d = 0; d < 10; ++d) hout[t * 10 + d] = fmaxf(hd[d], 0.0f);
}

// ---------------- Prep: zero agg/cnt, invn0 = 1/||x0|| ----------------
__global__ __launch_bounds__(256) void k_prep(
    const float* __restrict__ x0, float* __restrict__ agg,
    float* __restrict__ cnt, float* __restrict__ invn0)
{
    int j = blockIdx.x * 256 + threadIdx.x;
    if (j >= N0) return;
    float s = 0.0f;
#pragma unroll
    for (int i = 0; i < 10; ++i) {
        float v = x0[j * 10 + i];
        s += v * v;
        agg[j * 10 + i] = 0.0f;
    }
    cnt[j] = 0.0f;
    invn0[j] = 1.0f / sqrtf(s);
}

// ---------------- Scatter-add over base edges ----------------
__global__ __launch_bounds__(256) void k_scatter(
    const int* __restrict__ edges, const float* __restrict__ x0,
    float* __restrict__ agg, float* __restrict__ cnt)
{
    int e = blockIdx.x * 256 + threadIdx.x;
    if (e >= E0) return;
    int s = edges[e];
    int d = edges[E0 + e];
    atomicAdd(&cnt[d], 1.0f);
#pragma unroll
    for (int i = 0; i < 10; ++i) atomicAdd(&agg[d * 10 + i], x0[s * 10 + i]);
}

// ---------------- SAGE layer 1 on base nodes -> o1base[N0,15] ----------------
__global__ __launch_bounds__(256) void k_o1base(
    const float* __restrict__ x0, const float* __restrict__ agg,
    const float* __restrict__ cnt, const float* __restrict__ c1_wl,
    const float* __restrict__ c1_bl, const float* __restrict__ c1_wr,
    float* __restrict__ o1b)
{
    int j = blockIdx.x * 256 + threadIdx.x;
    if (j >= N0) return;
    float c = fmaxf(cnt[j], 1.0f);
    float mean[10], xr[10];
#pragma unroll
    for (int i = 0; i < 10; ++i) {
        mean[i] = agg[j * 10 + i] / c;
        xr[i] = x0[j * 10 + i];
    }
#pragma unroll
    for (int k = 0; k < 15; ++k) {
        float v = c1_bl[k];
#pragma unroll
        for (int i = 0; i < 10; ++i)
            v += mean[i] * c1_wl[k * 10 + i] + xr[i] * c1_wr[k * 10 + i];
        o1b[j * 15 + k] = fmaxf(v, 0.0f);
    }
}

// top-6 sorted-insert (registers only, constant indices)
#define TOP6_INSERT(UVAL, JJ)                                                         \
    if ((UVAL) > tv5) {                                                               \
        tv5 = (UVAL); ti5 = (JJ);                                                     \
        if (tv5 > tv4) { float f_=tv4; tv4=tv5; tv5=f_; int g_=ti4; ti4=ti5; ti5=g_; }\
        if (tv4 > tv3) { float f_=tv3; tv3=tv4; tv4=f_; int g_=ti3; ti3=ti4; ti4=g_; }\
        if (tv3 > tv2) { float f_=tv2; tv2=tv3; tv3=f_; int g_=ti2; ti2=ti3; ti3=g_; }\
        if (tv2 > tv1) { float f_=tv1; tv1=tv2; tv2=f_; int g_=ti1; ti1=ti2; ti2=g_; }\
        if (tv1 > tv0) { float f_=tv0; tv0=tv1; tv1=f_; int g_=ti0; ti0=ti1; ti1=g_; }\
    }

// ---------------- Main: WMMA cosine scores + top-6 + tail ----------------
__global__ __launch_bounds__(128) void k_main(
    const float* __restrict__ h, const float* __restrict__ x0,
    const float* __restrict__ invn0g, const float* __restrict__ o1base,
    const float* __restrict__ c1_bl, const float* __restrict__ c1_wr,
    const float* __restrict__ c2_wl, const float* __restrict__ c2_bl,
    const float* __restrict__ c2_wr, const float* __restrict__ lin_w,
    const float* __restrict__ lin_b, float* __restrict__ out)
{
    __shared__ float s_invn0[N0];
    // row stride 20 floats = 80B: 16B-aligned rows for b128 reads,
    // 16 distinct start banks (gcd(20,64)=4) -> conflict-free
    __shared__ float s_sc[4][16][20];

    for (int i = threadIdx.x; i < N0; i += 128) s_invn0[i] = invn0g[i];
    __syncthreads();   // only real block barrier: s_invn0 is cross-wave

    const int wave = threadIdx.x >> 5;
    const int lane = threadIdx.x & 31;
    const int m = lane & 15;
    const bool hiHalf = lane >= 16;
    const int tb = (blockIdx.x * 4 + wave) * 16;
    const int t = tb + m;

    // h row (10 floats, 8B-aligned: stride 40B)
    float hv[10];
    {
        const float2* hp = (const float2*)(h + t * 10);
        float2 q0 = hp[0], q1 = hp[1], q2 = hp[2], q3 = hp[3], q4 = hp[4];
        hv[0] = q0.x; hv[1] = q0.y; hv[2] = q1.x; hv[3] = q1.y; hv[4] = q2.x;
        hv[5] = q2.y; hv[6] = q3.x; hv[7] = q3.y; hv[8] = q4.x; hv[9] = q4.y;
    }
    float ssum = 0.0f;
#pragma unroll
    for (int k = 0; k < 10; ++k) ssum += hv[k] * hv[k];
    const float invnh = 1.0f / sqrtf(ssum);

    // A fragments (16x4 f32): lanes 0-15 hold K={4kc,4kc+1}, lanes 16-31 K={4kc+2,4kc+3}
    v2f a0, a1, a2;
    a0.x = hiHalf ? hv[2] : hv[0];
    a0.y = hiHalf ? hv[3] : hv[1];
    a1.x = hiHalf ? hv[6] : hv[4];
    a1.y = hiHalf ? hv[7] : hv[5];
    a2.x = hiHalf ? 0.0f : hv[8];   // K=10,11 zero padding
    a2.y = hiHalf ? 0.0f : hv[9];

    // top-6 of u = dot * invn0[j]  (invnh > 0 is rank-invariant per row)
    float tv0 = -INFINITY, tv1 = -INFINITY, tv2 = -INFINITY,
          tv3 = -INFINITY, tv4 = -INFINITY, tv5 = -INFINITY;
    int ti0 = 0, ti1 = 0, ti2 = 0, ti3 = 0, ti4 = 0, ti5 = 0;

    const int NFULL = (N0 / 16) * 16;   // 3184

    for (int jb = 0; jb < NFULL; jb += 16) {
        const int j = jb + m;
        const float2* xp = (const float2*)(x0 + j * 10);
        float2 p0 = xp[0], p1 = xp[1], p2 = xp[2], p3 = xp[3], p4 = xp[4];

        v2f b0, b1, b2;
        b0.x = hiHalf ? p1.x : p0.x;
        b0.y = hiHalf ? p1.y : p0.y;
        b1.x = hiHalf ? p3.x : p2.x;
        b1.y = hiHalf ? p3.y : p2.y;
        b2.x = hiHalf ? 0.0f : p4.x;
        b2.y = hiHalf ? 0.0f : p4.y;

        v8f c = {};
        c = __builtin_amdgcn_wmma_f32_16x16x4_f32(false, a0, false, b0,
                                                  (short)0, c, false, false);
        c = __builtin_amdgcn_wmma_f32_16x16x4_f32(false, a1, false, b1,
                                                  (short)0, c, false, false);
        c = __builtin_amdgcn_wmma_f32_16x16x4_f32(false, a2, false, b2,
                                                  (short)0, c, false, false);

        // C layout: lane holds col n = lane&15, rows r + (hiHalf?8:0);
        // fold column scale 1/||x0_j|| at spill time
        const float colscale = s_invn0[j];
#pragma unroll
        for (int r = 0; r < 8; ++r)
            s_sc[wave][r + (hiHalf ? 8 : 0)][m] = c[r] * colscale;
        __builtin_amdgcn_wave_barrier();   // wave-internal exchange only

        // read my row back as 4x b128
        const float4* rp = (const float4*)(&s_sc[wave][m][0]);
        float4 r0 = rp[0], r1 = rp[1], r2 = rp[2], r3 = rp[3];
        __builtin_amdgcn_wave_barrier();

        float mxA = fmaxf(fmaxf(r0.x, r0.y), fmaxf(r0.z, r0.w));
        float mxB = fmaxf(fmaxf(r1.x, r1.y), fmaxf(r1.z, r1.w));
        float mxC = fmaxf(fmaxf(r2.x, r2.y), fmaxf(r2.z, r2.w));
        float mxD = fmaxf(fmaxf(r3.x, r3.y), fmaxf(r3.z, r3.w));
        float rowmax = fmaxf(fmaxf(mxA, mxB), fmaxf(mxC, mxD));

        if (rowmax > tv5) {   // rare once top-6 is warm
            TOP6_INSERT(r0.x, jb + 0);  TOP6_INSERT(r0.y, jb + 1);
            TOP6_INSERT(r0.z, jb + 2);  TOP6_INSERT(r0.w, jb + 3);
            TOP6_INSERT(r1.x, jb + 4);  TOP6_INSERT(r1.y, jb + 5);
            TOP6_INSERT(r1.z, jb + 6);  TOP6_INSERT(r1.w, jb + 7);
            TOP6_INSERT(r2.x, jb + 8);  TOP6_INSERT(r2.y, jb + 9);
            TOP6_INSERT(r2.z, jb + 10); TOP6_INSERT(r2.w, jb + 11);
            TOP6_INSERT(r3.x, jb + 12); TOP6_INSERT(r3.y, jb + 13);
            TOP6_INSERT(r3.z, jb + 14); TOP6_INSERT(r3.w, jb + 15);
        }
    }

    // tail tile: columns NFULL .. N0-1 (6 valid)
    {
        int j = NFULL + m;
        if (j >= N0) j = N0 - 1;            // clamp address; extra cols ignored
        const float2* xp = (const float2*)(x0 + j * 10);
        float2 p0 = xp[0], p1 = xp[1], p2 = xp[2], p3 = xp[3], p4 = xp[4];

        v2f b0, b1, b2;
        b0.x = hiHalf ? p1.x : p0.x;
        b0.y = hiHalf ? p1.y : p0.y;
        b1.x = hiHalf ? p3.x : p2.x;
        b1.y = hiHalf ? p3.y : p2.y;
        b2.x = hiHalf ? 0.0f : p4.x;
        b2.y = hiHalf ? 0.0f : p4.y;

        v8f c = {};
        c = __builtin_amdgcn_wmma_f32_16x16x4_f32(false, a0, false, b0,
                                                  (short)0, c, false, false);
        c = __builtin_amdgcn_wmma_f32_16x16x4_f32(false, a1, false, b1,
                                                  (short)0, c, false, false);
        c = __builtin_amdgcn_wmma_f32_16x16x4_f32(false, a2, false, b2,
                                                  (short)0, c, false, false);

        const float colscale = s_invn0[j];
#pragma unroll
        for (int r = 0; r < 8; ++r)
            s_sc[wave][r + (hiHalf ? 8 : 0)][m] = c[r] * colscale;
        __builtin_amdgcn_wave_barrier();

        const float4* rp = (const float4*)(&s_sc[wave][m][0]);
        float4 r0 = rp[0];
        float2 r4 = *(const float2*)(&s_sc[wave][m][4]);
        __builtin_amdgcn_wave_barrier();

        TOP6_INSERT(r0.x, NFULL + 0);
        TOP6_INSERT(r0.y, NFULL + 1);
        TOP6_INSERT(r0.z, NFULL + 2);
        TOP6_INSERT(r0.w, NFULL + 3);
        TOP6_INSERT(r4.x, NFULL + 4);
        TOP6_INSERT(r4.y, NFULL + 5);
    }

    if (!hiHalf) {
        // apply row scale only for the exact self-match test (rank-invariant)
        bool selfm = ((tv0 * invnh) == 1.0f);
        int id[5];
        id[0] = selfm ? ti1 : ti0;
        id[1] = selfm ? ti2 : ti1;
        id[2] = selfm ? ti3 : ti2;
        id[3] = selfm ? ti4 : ti3;
        id[4] = selfm ? ti5 : ti4;

        float aggv[15];
#pragma unroll
        for (int k = 0; k < 15; ++k) aggv[k] = 0.0f;
#pragma unroll
        for (int e = 0; e < 5; ++e) {
            const float* row = o1base + id[e] * 15;
#pragma unroll
            for (int k = 0; k < 15; ++k) aggv[k] += row[k];
        }
#pragma unroll
        for (int k = 0; k < 15; ++k) aggv[k] /= 5.0f;

        // o1 for test node: relu(c1_bl + h @ c1_wr.T)  (no in-edges in base graph)
        float o1t[15];
#pragma unroll
        for (int k = 0; k < 15; ++k) {
            float v = c1_bl[k];
#pragma unroll
            for (int d = 0; d < 10; ++d) v += hv[d] * c1_wr[k * 10 + d];
            o1t[k] = fmaxf(v, 0.0f);
        }
        float o2[10];
#pragma unroll
        for (int d = 0; d < 10; ++d) {
            float v = c2_bl[d];
#pragma unroll
            for (int k = 0; k < 15; ++k)
                v += aggv[k] * c2_wl[d * 15 + k] + o1t[k] * c2_wr[d * 15 + k];
            o2[d] = v;
        }
        float l0 = lin_b[0], l1 = lin_b[1], l2 = lin_b[2];
#pragma unroll
        for (int d = 0; d < 10; ++d) {
            l0 += o2[d] * lin_w[0 * 10 + d];
            l1 += o2[d] * lin_w[1 * 10 + d];
            l2 += o2[d] * lin_w[2 * 10 + d];
        }
        float mx = fmaxf(l0, fmaxf(l1, l2));
        float e0 = expf(l0 - mx), e1 = expf(l1 - mx), e2 = expf(l2 - mx);
        float inv = 1.0f / (e0 + e1 + e2);
        out[t * 3 + 0] = e0 * inv;
        out[t * 3 + 1] = e1 * inv;
        out[t * 3 + 2] = e2 * inv;
    }
}

extern "C" void kernel_launch(void* const* d_in, const int* in_sizes, int n_in,
                              void* d_out, int out_size, void* d_ws, size_t ws_size,
                              hipStream_t stream) {
    (void)in_sizes; (void)n_in; (void)out_size; (void)ws_size;
    const float* x     = (const float*)d_in[0];
    const float* x0    = (const float*)d_in[1];
    const int*   edges = (const int*)d_in[2];
    const float* emb_w = (const float*)d_in[3];
    const float* fc_w  = (const float*)d_in[4];
    const float* fc_b  = (const float*)d_in[5];
    const float* fc2_w = (const float*)d_in[6];
    const float* fc2_b = (const float*)d_in[7];
    const float* c1_wl = (const float*)d_in[8];
    const float* c1_bl = (const float*)d_in[9];
    const float* c1_wr = (const float*)d_in[10];
    const float* c2_wl = (const float*)d_in[11];
    const float* c2_bl = (const float*)d_in[12];
    const float* c2_wr = (const float*)d_in[13];
    const float* lin_w = (const float*)d_in[14];
    const float* lin_b = (const float*)d_in[15];
    float* out = (float*)d_out;

    float* ws    = (float*)d_ws;
    float* h     = ws;                 // NT*10
    float* agg   = h + NT * 10;        // N0*10
    float* cnt   = agg + N0 * 10;      // N0
    float* invn0 = cnt + N0;           // N0
    float* o1b   = invn0 + N0;         // N0*15

    k_encoder<<<(NT + 255) / 256, 256, 0, stream>>>(x, emb_w, fc_w, fc_b,
                                                    fc2_w, fc2_b, h);
    k_prep<<<(N0 + 255) / 256, 256, 0, stream>>>(x0, agg, cnt, invn0);
    k_scatter<<<(E0 + 255) / 256, 256, 0, stream>>>(edges, x0, agg, cnt);
    k_o1base<<<(N0 + 255) / 256, 256, 0, stream>>>(x0, agg, cnt, c1_wl, c1_bl,
                                                   c1_wr, o1b);
    k_main<<<NT / 64, 128, 0, stream>>>(h, x0, invn0, o1b, c1_bl, c1_wr, c2_wl,
                                        c2_bl, c2_wr, lin_w, lin_b, out);
}